// SpatiallyVariantConvolution_47708496724546
// MI455X (gfx1250) — compile-verified
//
#include <hip/hip_runtime.h>

// Spatially-variant 5x5 convolution for MI455X (gfx1250, wave32).
//
// Compute: D = A x B + C with V_WMMA_F32_16X16X4_F32 (full fp32, K=4).
//   A[c,(i,u)]  : LDS-staged padded features (16 channels x 100 K-entries)
//   B[(i,u),xr] : banded per-pixel kernel matrix, kern[i, u-xr] for 0<=u-xr<5
//   D[c,xr]     : 16x16 output tile, 25 K=4 WMMA steps, B shared by both
//                 channel groups (kernels are channel-invariant).
//
// Data movement: memory-bound problem (~4.5 FLOP/B, 47 MB working set, fits
// in the 192 MB L2). Staging uses CDNA5 async copies
// (GLOBAL_LOAD_ASYNC_TO_LDS_B64/B128, ASYNCcnt + s_wait_asynccnt): bytes go
// HBM/L2 -> LDS with no VGPR round-trip; 25-tap spatial reuse and the 32x
// channel reuse of kernels are then served from LDS.

typedef float v2f __attribute__((ext_vector_type(2)));
typedef float v8f __attribute__((ext_vector_type(8)));

#define BATCH 2
#define CHAN  32
#define HH    256
#define WW    256
#define KS    5
#define PADR  2

#define XSPAN   64                  // x-columns per workgroup
#define WAVES   4                   // one 16-wide x-tile per wave
#define THREADS (WAVES * 32)
#define FCOLS   (XSPAN + 2 * PADR)  // 68 padded feature columns in LDS
#define FROWS   (CHAN * KS)         // 160 (channel, tap-row) rows in LDS
#define FCH_STRIDE (KS * FCOLS)     // 340 dwords: 8B-aligned async dests;
                                    // 340 % 64 = 20 -> conflict-free gathers
#define KROWS   (KS * KS)           // 25

// ---- CDNA5 async memory->LDS copies (tracked on ASYNCcnt) ----
__device__ __forceinline__ void async_ld_b64(unsigned lds_off, const float* g) {
    asm volatile("global_load_async_to_lds_b64 %0, %1, off"
                 :: "v"(lds_off), "v"(g) : "memory");
}
__device__ __forceinline__ void async_ld_b128(unsigned lds_off, const float* g) {
    asm volatile("global_load_async_to_lds_b128 %0, %1, off"
                 :: "v"(lds_off), "v"(g) : "memory");
}
__device__ __forceinline__ void wait_asynccnt0() {
#if __has_builtin(__builtin_amdgcn_s_wait_asynccnt)
    __builtin_amdgcn_s_wait_asynccnt(0);
#else
    asm volatile("s_wait_asynccnt 0x0" ::: "memory");
#endif
}

__global__ __launch_bounds__(THREADS)
void svconv_wmma_f32(const float* __restrict__ kern,
                     const float* __restrict__ feat,
                     float* __restrict__ out)
{
    __shared__ __align__(16) float ldsF[CHAN * FCH_STRIDE];  // 43,520 B
    __shared__ __align__(16) float ldsK[KROWS * XSPAN];      //  6,400 B

    const int bid = blockIdx.x;
    const int xb  = (bid & 3) * XSPAN;         // block's first x column
    const int y   = (bid >> 2) & (HH - 1);     // output row
    const int b   = bid >> 10;                 // batch (/(4*256))

    const int tid = threadIdx.x;

    // ---- kernels: 25 taps x 64 cols, fully aligned -> async B128 ----
    for (int ch = tid; ch < KROWS * 16; ch += THREADS) {
        const int w4 = ch & 15;                // 16B chunk within tap row
        const int ij = ch >> 4;                // tap index 0..24
        const unsigned lo = (unsigned)(size_t)&ldsK[ij * XSPAN + 4 * w4];
        async_ld_b128(lo, kern + ((size_t)(b * KROWS + ij) * HH + y) * WW
                              + xb + 4 * w4);
    }

    // ---- features, interior: 160 rows x 32 async B64 chunks ----
    for (int ch = tid; ch < FROWS * 32; ch += THREADS) {
        const int k2 = ch & 31;                // 8B chunk within row
        const int r  = ch >> 5;                // 0..159
        const int i  = r % KS;
        const int c  = r / KS;
        const int row = y + i - PADR;
        const int fo  = c * FCH_STRIDE + i * FCOLS + 2 + 2 * k2;  // even dw
        if ((unsigned)row < (unsigned)HH) {
            async_ld_b64((unsigned)(size_t)&ldsF[fo],
                         feat + ((size_t)(b * CHAN + c) * HH + row) * WW
                              + xb + 2 * k2);
        } else {
            *(float2*)&ldsF[fo] = make_float2(0.0f, 0.0f);        // zero halo row
        }
    }

    // ---- features, halo columns u in {0,1,66,67}: predicated zero-fill ----
    for (int e = tid; e < FROWS * 4; e += THREADS) {
        const int hu = e & 3;
        const int r  = e >> 2;
        const int i  = r % KS;
        const int c  = r / KS;
        const int u  = (hu < 2) ? hu : (XSPAN + hu);   // 0,1,66,67
        const int row = y + i - PADR;
        const int col = xb + u - PADR;
        float v = 0.0f;
        if ((unsigned)row < (unsigned)HH && (unsigned)col < (unsigned)WW)
            v = feat[((size_t)(b * CHAN + c) * HH + row) * WW + col];
        ldsF[c * FCH_STRIDE + i * FCOLS + u] = v;
    }

    wait_asynccnt0();          // drain this wave's async copies
    __syncthreads();           // make LDS tile visible to all waves

    const int wave = tid >> 5;
    const int lane = tid & 31;
    const int L    = lane & 15;    // M (channel) for A, N (column) for B/D
    const int half = lane >> 4;    // selects K pair (A/B); M 0-7 vs 8-15 (D)
    const int xt   = wave * 16;    // wave's x-tile offset inside block span

    v8f acc0 = {};                 // channels 0..15
    v8f acc1 = {};                 // channels 16..31

    const float* fbase0 = &ldsF[(0 * 16 + L) * FCH_STRIDE];
    const float* fbase1 = &ldsF[(1 * 16 + L) * FCH_STRIDE];

    // K' = 100 = 25 chunks of K=4.  t = i*20 + u,  u = xr + j in [0,20).
#pragma unroll
    for (int q = 0; q < 25; ++q) {
        const int t0 = 4 * q + 2 * half;   // K index of the .x element (even)
        const int i0 = t0 / 20;
        const int u0 = t0 % 20;            // even -> pair stays in one tap row

        // A fragment (ISA 16x4 f32 layout): lane L = row M,
        //   VGPR0 = K = 2*half, VGPR1 = K = 2*half+1 within the chunk.
        const int fo = i0 * FCOLS + xt + u0;
        v2f a0, a1, bb;
        a0.x = fbase0[fo];  a0.y = fbase0[fo + 1];
        a1.x = fbase1[fo];  a1.y = fbase1[fo + 1];

        // B fragment (mirrored layout): lane L = column N, banded kernel row.
        const int j0 = u0 - L;
        const int j1 = j0 + 1;
        bb.x = ((unsigned)j0 < KS) ? ldsK[(i0 * KS + j0) * XSPAN + xt + L] : 0.0f;
        bb.y = ((unsigned)j1 < KS) ? ldsK[(i0 * KS + j1) * XSPAN + xt + L] : 0.0f;

        acc0 = __builtin_amdgcn_wmma_f32_16x16x4_f32(
                   false, a0, false, bb, (short)0, acc0, false, false);
        acc1 = __builtin_amdgcn_wmma_f32_16x16x4_f32(
                   false, a1, false, bb, (short)0, acc1, false, false);
    }

    // ---- store D: lane L = column, VGPR v = row (M = v + 8*half) ----
    const int xcol = xb + xt + L;
#pragma unroll
    for (int v = 0; v < 8; ++v) {
        const int m = v + 8 * half;
        out[((size_t)(b * CHAN + m) * HH + y) * WW + xcol]      = acc0[v];
        out[((size_t)(b * CHAN + 16 + m) * HH + y) * WW + xcol] = acc1[v];
    }
}

extern "C" void kernel_launch(void* const* d_in, const int* in_sizes, int n_in,
                              void* d_out, int out_size, void* d_ws, size_t ws_size,
                              hipStream_t stream) {
    (void)in_sizes; (void)n_in; (void)d_ws; (void)ws_size; (void)out_size;
    const float* kernels  = (const float*)d_in[0];  // (2,5,5,256,256) f32
    const float* features = (const float*)d_in[1];  // (2,32,256,256) f32
    float* out = (float*)d_out;                     // (2,32,256,256) f32

    const int blocks = BATCH * HH * (WW / XSPAN);   // 2*256*4 = 2048
    svconv_wmma_f32<<<blocks, THREADS, 0, stream>>>(kernels, features, out);
}